// Poi2d_with_a_exact_2559800508490
// MI455X (gfx1250) — compile-verified
//
#include <hip/hip_runtime.h>

// out[i] = 0.5 * a[0] * x0*(x0-1) * x1*(x1-1), input interleaved (x0,x1) pairs.
//
// Memory-bound streaming kernel for MI455X (gfx1250, wave32):
//  - each thread: 2x global_load_b128 (4 point pairs) -> 1x global_store_b128 (4 outputs)
//  - input loads RT (134 MB fits the 192 MB L2 -> L2-resident across graph replays)
//  - output stores TH_NT (write-once stream, don't evict input from L2)

typedef float v4f __attribute__((ext_vector_type(4)));

__global__ __launch_bounds__(256) void poi2d_vec4_kernel(
    const v4f* __restrict__ in4,      // N/2 v4f: {x0,x1,x0',x1'}
    const float* __restrict__ a,
    v4f* __restrict__ out4,           // N/4 v4f
    int n4)                           // N/4
{
    int i = blockIdx.x * blockDim.x + threadIdx.x;
    if (i >= n4) return;

    const float s = 0.5f * a[0];      // uniform: s_load_b32 + v_mul

    // two contiguous 16B loads -> 4 (x0,x1) pairs
    v4f u = in4[2 * i + 0];           // pairs 0,1
    v4f v = in4[2 * i + 1];           // pairs 2,3

    // p = x*(x-1) = fma(x, x, -x)
    float p0a = __builtin_fmaf(u.x, u.x, -u.x);
    float p1a = __builtin_fmaf(u.y, u.y, -u.y);
    float p0b = __builtin_fmaf(u.z, u.z, -u.z);
    float p1b = __builtin_fmaf(u.w, u.w, -u.w);
    float p0c = __builtin_fmaf(v.x, v.x, -v.x);
    float p1c = __builtin_fmaf(v.y, v.y, -v.y);
    float p0d = __builtin_fmaf(v.z, v.z, -v.z);
    float p1d = __builtin_fmaf(v.w, v.w, -v.w);

    v4f r;
    r.x = s * p0a * p1a;
    r.y = s * p0b * p1b;
    r.z = s * p0c * p1c;
    r.w = s * p0d * p1d;

    // non-temporal 16B store: global_store_b128 th:TH_NT
    __builtin_nontemporal_store(r, &out4[i]);
}

// scalar tail (not exercised for N = 2^24, kept for generality)
__global__ __launch_bounds__(64) void poi2d_tail_kernel(
    const float* __restrict__ in,
    const float* __restrict__ a,
    float* __restrict__ out,
    int start, int n)
{
    int i = start + blockIdx.x * blockDim.x + threadIdx.x;
    if (i >= n) return;
    float s  = 0.5f * a[0];
    float x0 = in[2 * i + 0];
    float x1 = in[2 * i + 1];
    float p0 = __builtin_fmaf(x0, x0, -x0);
    float p1 = __builtin_fmaf(x1, x1, -x1);
    __builtin_nontemporal_store(s * p0 * p1, &out[i]);
}

extern "C" void kernel_launch(void* const* d_in, const int* in_sizes, int n_in,
                              void* d_out, int out_size, void* d_ws, size_t ws_size,
                              hipStream_t stream) {
    const float* in = (const float*)d_in[0];   // (N,2) f32, interleaved
    const float* a  = (const float*)d_in[1];   // (1,)  f32
    float* out = (float*)d_out;                // (N,1) f32

    const int n  = out_size;                   // N = 16,777,216
    const int n4 = n >> 2;                     // outputs per-thread = 4

    if (n4 > 0) {
        const int block = 256;                 // 8 wave32s per block
        const int grid  = (n4 + block - 1) / block;
        poi2d_vec4_kernel<<<grid, block, 0, stream>>>(
            (const v4f*)in, a, (v4f*)out, n4);
    }

    const int done = n4 << 2;
    if (done < n) {
        const int rem   = n - done;
        const int block = 64;
        const int grid  = (rem + block - 1) / block;
        poi2d_tail_kernel<<<grid, block, 0, stream>>>(in, a, out, done, n);
    }
}